// Top2Gating_77214922047856
// MI455X (gfx1250) — compile-verified
//
#include <hip/hip_runtime.h>
#include <hip/hip_bf16.h>
#include <math.h>

typedef __attribute__((ext_vector_type(2))) float v2f;
typedef __attribute__((ext_vector_type(8))) float v8f;

#define BATCH 8
#define NTOKS 2048
#define DMODEL 1024
#define NEXP 64
#define CAP 40
#define NTOT (BATCH * NTOKS)                 // 16384 tokens
static const size_t N4 = (size_t)NTOT * NEXP * CAP;  // 41,943,040 elems per tensor

// ---- workspace layout (4-byte units) ----
#define WS_IDX1   0
#define WS_IDX2   (NTOT)
#define WS_G1     (2 * NTOT)
#define WS_G2     (3 * NTOT)
#define WS_SLOT1  (4 * NTOT)
#define WS_SLOT2  (5 * NTOT)
#define WS_CNT1   (6 * NTOT)                 // 512 ints
#define WS_RAW    (6 * NTOT + 512)           // 512 floats (density proxy sums)
#define WS_ZSUM   (6 * NTOT + 1024)          // 1 float
// total ~398 KB of d_ws

// ---------------------------------------------------------------------------
// Stage 1: logits GEMM (fp32 WMMA) + softmax + top2 + routing, fused per
// 16-token tile. Block = 128 threads = 4 wave32s; wave w owns experts
// [16w, 16w+16). K=1024 consumed in 8 chunks of 128 staged in LDS.
// ---------------------------------------------------------------------------
__global__ __launch_bounds__(128)
void gate_kernel(const float* __restrict__ x, const float* __restrict__ w,
                 const float* __restrict__ noise,
                 int* __restrict__ idx1, int* __restrict__ idx2r,
                 float* __restrict__ g1a, float* __restrict__ g2a,
                 int* __restrict__ slot1, int* __restrict__ slot2,
                 float* __restrict__ rawsum, float* __restrict__ zsum)
{
    __shared__ float ldsA[16 * 129];   // padded: stride 129 -> conflict-free
    __shared__ float ldsL[16 * 65];    // 16 tokens x 64 expert logits
    __shared__ float mxs[16], sms[16], zpart[16], dpart[128];

    const int tid  = threadIdx.x;
    const int wave = tid >> 5;
    const int lane = tid & 31;
    const int half = lane >> 4;        // 0: K=0,1  1: K=2,3 (A/B frag layout)
    const int l16  = lane & 15;
    const int koff = half << 1;
    const int tok0 = blockIdx.x << 4;  // 16 tokens per block

    v8f acc = {};

    for (int kb = 0; kb < 8; ++kb) {
        const int kBase = kb << 7;
        // cooperative coalesced load of 16x128 A-chunk (float4)
        for (int i = tid; i < 512; i += 128) {
            const int r  = i >> 5;
            const int c4 = (i & 31) << 2;
            const float4 v = *(const float4*)(x + (size_t)(tok0 + r) * DMODEL + kBase + c4);
            float* dst = &ldsA[r * 129 + c4];
            dst[0] = v.x; dst[1] = v.y; dst[2] = v.z; dst[3] = v.w;
        }
        __syncthreads();
        const float* wp0 = w + (size_t)(kBase + koff) * NEXP + (wave << 4) + l16;
        for (int kk = 0; kk < 128; kk += 4) {
            v2f a, bf;
            a.x = ldsA[l16 * 129 + kk + koff];
            a.y = ldsA[l16 * 129 + kk + koff + 1];
            const float* wp = wp0 + (size_t)kk * NEXP;
            bf.x = wp[0];
            bf.y = wp[NEXP];
            acc = __builtin_amdgcn_wmma_f32_16x16x4_f32(
                false, a, false, bf, (short)0, acc, false, false);
        }
        __syncthreads();
    }

    // spill logits tile to LDS: vgpr r of lane -> token m = r + 8*half, expert = 16w + l16
    for (int r = 0; r < 8; ++r) {
        const int m = r + (half << 3);
        ldsL[m * 65 + (wave << 4) + l16] = acc[r];
    }
    __syncthreads();

    // per-token softmax / top2 / normalize / stochastic route (threads 0..15)
    if (tid < 16) {
        const float* row = &ldsL[tid * 65];
        float mx = row[0]; int i1 = 0;
        for (int e = 1; e < NEXP; ++e) { float v = row[e]; if (v > mx) { mx = v; i1 = e; } }
        float v2 = -3.4e38f; int i2 = 0;
        for (int e = 0; e < NEXP; ++e) {
            if (e == i1) continue;
            float v = row[e]; if (v > v2) { v2 = v; i2 = e; }
        }
        float s = 0.f;
        for (int e = 0; e < NEXP; ++e) s += __expf(row[e] - mx);
        const float g1 = 1.0f / s;                 // exp(mx-mx)/s
        const float g2 = __expf(v2 - mx) / s;
        const float den = g1 + g2 + 1e-9f;
        const float g1n = g1 / den, g2n = g2 / den;
        const size_t gt = (size_t)tok0 + tid;
        const bool rt = noise[gt] < (g2n * 5.0f);  // g2n / SECOND_THRESHOLD(0.2)
        idx1[gt]  = i1;
        idx2r[gt] = rt ? i2 : -1;
        g1a[gt] = g1n;
        g2a[gt] = g2n;
        slot1[gt] = -1;
        slot2[gt] = -1;
        const float lse = mx + __logf(s);
        zpart[tid] = lse * lse;
        mxs[tid] = mx; sms[tid] = s;
    }
    __syncthreads();

    // density proxy (sum of raw softmax probs over tokens) per expert
    const int e  = tid & 63;
    const int m0 = (tid >> 6) << 3;
    float part = 0.f;
    for (int m = m0; m < m0 + 8; ++m)
        part += __expf(ldsL[m * 65 + e] - mxs[m]) / sms[m];
    dpart[tid] = part;
    __syncthreads();
    const int bIdx = blockIdx.x >> 7;   // 128 tiles per batch
    if (tid < 64)
        atomicAdd(&rawsum[bIdx * NEXP + tid], dpart[tid] + dpart[tid + 64]);
    if (tid == 0) {
        float z = 0.f;
        for (int i = 0; i < 16; ++i) z += zpart[i];
        atomicAdd(zsum, z);
    }
}

// ---------------------------------------------------------------------------
// Stage 2: per-(b,e) exclusive cumsum over tokens via wave32 ballot scan.
// One wave per (b,e); 512 waves total. Capacity check pos < 40.
// ---------------------------------------------------------------------------
__global__ __launch_bounds__(256)
void scan_kernel(const int* __restrict__ idx1, const int* __restrict__ idx2r,
                 int* __restrict__ slot1, int* __restrict__ slot2,
                 int* __restrict__ cnt1)
{
    const int w    = (blockIdx.x << 3) + (threadIdx.x >> 5);  // 0..511
    const int lane = threadIdx.x & 31;
    const int b = w >> 6, e = w & 63;
    const int base = b * NTOKS;
    const unsigned below = (1u << lane) - 1u;
    unsigned run1 = 0, run2 = 0;
    for (int c = 0; c < NTOKS; c += 32) {
        const int t = base + c + lane;
        const bool m1 = (idx1[t] == e);
        const unsigned bal1 = __builtin_amdgcn_ballot_w32(m1);
        if (m1) {
            const unsigned p = (unsigned)__builtin_popcount(bal1 & below) + run1;
            slot1[t] = (p < CAP) ? (int)p : -1;
        }
        run1 += (unsigned)__builtin_popcount(bal1);
        const bool m2 = (idx2r[t] == e);
        const unsigned bal2 = __builtin_amdgcn_ballot_w32(m2);
        if (m2) {
            const unsigned p = (unsigned)__builtin_popcount(bal2 & below) + run2;
            slot2[t] = (p < CAP) ? (int)p : -1;
        }
        run2 += (unsigned)__builtin_popcount(bal2);
    }
    if (lane == 0) cnt1[w] = (int)run1;   // pre-capacity count (density_1)
}

// ---------------------------------------------------------------------------
// Stage 4: sparse scatter of combine/dispatch non-zeros (<=2 per token).
// ---------------------------------------------------------------------------
__global__ __launch_bounds__(256)
void scatter_kernel(const int* __restrict__ idx1, const int* __restrict__ idx2r,
                    const float* __restrict__ g1a, const float* __restrict__ g2a,
                    const int* __restrict__ slot1, const int* __restrict__ slot2,
                    float* __restrict__ out, size_t n4)
{
    const int t = blockIdx.x * 256 + threadIdx.x;
    if (t >= NTOT) return;
    float* dispatch = out;
    float* combine  = out + n4;
    const int s1 = slot1[t];
    if (s1 >= 0) {
        const size_t off = ((size_t)t * NEXP + idx1[t]) * CAP + s1;
        combine[off]  = g1a[t];
        dispatch[off] = 1.0f;
    }
    const int s2 = slot2[t];
    if (s2 >= 0) {
        const size_t off = ((size_t)t * NEXP + idx2r[t]) * CAP + s2;
        combine[off]  = g2a[t];
        dispatch[off] = 1.0f;
    }
}

// ---------------------------------------------------------------------------
// Stage 5: losses. balance = sum_{b,e}(raw/n * cnt/n) * e*e / (b*e)
// ---------------------------------------------------------------------------
__global__ __launch_bounds__(512)
void finalize_kernel(const float* __restrict__ rawsum, const int* __restrict__ cnt1,
                     const float* __restrict__ zsum, float* __restrict__ out, size_t n4)
{
    __shared__ float red[512];
    const int i = threadIdx.x;
    red[i] = (rawsum[i] * (1.f / NTOKS)) * ((float)cnt1[i] * (1.f / NTOKS));
    __syncthreads();
    for (int s = 256; s > 0; s >>= 1) {
        if (i < s) red[i] += red[i + s];
        __syncthreads();
    }
    if (i == 0) {
        out[2 * n4]     = red[0] * ((float)(NEXP * NEXP) / (float)(BATCH * NEXP)); // *8
        out[2 * n4 + 1] = zsum[0] * (1.f / (float)NTOT);
    }
}

extern "C" void kernel_launch(void* const* d_in, const int* in_sizes, int n_in,
                              void* d_out, int out_size, void* d_ws, size_t ws_size,
                              hipStream_t stream) {
    const float* x     = (const float*)d_in[0];   // [8,2048,1024]
    const float* w     = (const float*)d_in[1];   // [1024,64]
    const float* noise = (const float*)d_in[2];   // [8,2048]
    float* out = (float*)d_out;

    int*   wsI = (int*)d_ws;
    float* wsF = (float*)d_ws;
    int*   idx1  = wsI + WS_IDX1;
    int*   idx2r = wsI + WS_IDX2;
    float* g1a   = wsF + WS_G1;
    float* g2a   = wsF + WS_G2;
    int*   slot1 = wsI + WS_SLOT1;
    int*   slot2 = wsI + WS_SLOT2;
    int*   cnt1  = wsI + WS_CNT1;
    float* raws  = wsF + WS_RAW;
    float* zsum  = wsF + WS_ZSUM;

    // zero the dense output tensors (bandwidth floor) + loss accumulators
    hipMemsetAsync(out, 0, 2 * N4 * sizeof(float), stream);
    hipMemsetAsync(raws, 0, 513 * sizeof(float), stream);

    gate_kernel<<<NTOT / 16, 128, 0, stream>>>(x, w, noise, idx1, idx2r,
                                               g1a, g2a, slot1, slot2, raws, zsum);
    scan_kernel<<<64, 256, 0, stream>>>(idx1, idx2r, slot1, slot2, cnt1);
    scatter_kernel<<<NTOT / 256, 256, 0, stream>>>(idx1, idx2r, g1a, g2a,
                                                   slot1, slot2, out, N4);
    finalize_kernel<<<1, 512, 0, stream>>>(raws, cnt1, zsum, out, N4);
}